// VectorQuantizer_70325794505310
// MI455X (gfx1250) — compile-verified
//
#include <hip/hip_runtime.h>
#include <hip/hip_bf16.h>

typedef __attribute__((ext_vector_type(2))) float v2f;
typedef __attribute__((ext_vector_type(4))) float v4f;
typedef __attribute__((ext_vector_type(8))) float v8f;

#define N_TOT   131072      // B*H*W
#define K_CODES 512
#define DDIM    64
#define HW      4096        // H*W
#define CHW     262144      // C*H*W

// ---------------------------------------------------------------------------
// ws layout (floats): [0,32768) Bpack | [32768,33280) enorm | [33280,33792) counts | [33792] loss
// d_out layout (floats): [0,8388608) quantized NCHW | [8388608] loss | [8388609] perplexity
//                        [8388610] usage | [8388611, +131072) indices (as float)
// ---------------------------------------------------------------------------

// Pack embedding into WMMA-B lane layout, compute ||e||^2, zero accumulators.
// Bpack element (nt, g, lane, comp): c = 2g + (comp>>1), vg = comp&1,
//   value = emb[nt*16 + (lane&15)][4c + 2*(lane>>4) + vg]
__global__ __launch_bounds__(256) void vq_prep(const float* __restrict__ emb,
                                               float* __restrict__ Bpack,
                                               float* __restrict__ enorm,
                                               float* __restrict__ counts,
                                               float* __restrict__ loss) {
    int e = blockIdx.x * 256 + threadIdx.x;        // 128 blocks -> 32768 threads
    if (e < 32768) {
        int nt = e >> 10, rem = e & 1023;
        int g = rem >> 7, lane = (rem >> 2) & 31, comp = rem & 3;
        int c = 2 * g + (comp >> 1), vg = comp & 1;
        int n = nt * 16 + (lane & 15);
        int d = 4 * c + 2 * (lane >> 4) + vg;
        Bpack[e] = emb[n * DDIM + d];
    }
    if (e < K_CODES) {
        float s = 0.0f;
        #pragma unroll 8
        for (int d = 0; d < DDIM; ++d) { float v = emb[e * DDIM + d]; s = fmaf(v, v, s); }
        enorm[e]  = s;
        counts[e] = 0.0f;
    }
    if (e == 0) *loss = 0.0f;
}

// One wave per 16-row tile of x vs all 512 codes via chained V_WMMA_F32_16X16X4_F32.
// Two independent accumulator chains (even/odd K-chunks) to halve WMMA RAW depth.
__global__ __launch_bounds__(256) void vq_main(const float* __restrict__ x,
                                               const float* __restrict__ Bpack,
                                               const float* __restrict__ enorm,
                                               float* __restrict__ counts,
                                               float* __restrict__ loss_sum,
                                               float* __restrict__ out_idx_f) {
    const int lane = threadIdx.x & 31;
    const int wave = threadIdx.x >> 5;
    const int tile = blockIdx.x * 8 + wave;         // 1024 blocks * 8 waves = 8192 tiles
    const int lm   = lane & 15;
    const int half = lane >> 4;

    // ---- Load A tile (16 rows x 64 dims) into 32 VGPRs, WMMA A layout ----
    // flat[n][d] with n=b*HW+h*W+w, d=c  ->  x[(n>>12)*CHW + d*HW + (n&4095)]
    const int n_row = tile * 16 + lm;
    const float* xr = x + (size_t)(n_row >> 12) * CHW + (n_row & (HW - 1));
    float a[32];
    float xnorm = 0.0f;
    #pragma unroll
    for (int c = 0; c < 16; ++c) {
        int d0 = 4 * c + 2 * half;
        float a0 = xr[(size_t)d0 * HW];
        float a1 = xr[(size_t)(d0 + 1) * HW];
        a[2 * c]     = a0;
        a[2 * c + 1] = a1;
        xnorm = fmaf(a0, a0, fmaf(a1, a1, xnorm));
    }
    xnorm += __shfl_xor(xnorm, 16, 32);             // full ||x_row||^2 (row = lm)

    float minv[8];
    int   mini[8];
    #pragma unroll
    for (int j = 0; j < 8; ++j) { minv[j] = 3.4e38f; mini[j] = 0; }

    // ---- Sweep 32 code tiles of 16 ----
    for (int nt = 0; nt < 32; ++nt) {
        if (nt < 31)
            __builtin_prefetch((const void*)(Bpack + (nt + 1) * 1024), 0, 3); // near-scope prefetch
        const v4f* bp = (const v4f*)(Bpack + nt * 1024);
        v4f q[8];
        #pragma unroll
        for (int g = 0; g < 8; ++g) q[g] = bp[g * 32 + lane];   // coalesced b128
        float en = enorm[nt * 16 + lm];

        v8f acc0 = {};
        v8f acc1 = {};
        #pragma unroll
        for (int c = 0; c < 16; c += 2) {
            // even chunk -> chain 0
            {
                v2f av = { a[2 * c], a[2 * c + 1] };
                v4f qq = q[c >> 1];
                v2f bv = { qq.x, qq.y };
                acc0 = __builtin_amdgcn_wmma_f32_16x16x4_f32(
                    false, av, false, bv, (short)0, acc0, false, false);
            }
            // odd chunk -> chain 1 (independent of chain 0)
            {
                v2f av = { a[2 * (c + 1)], a[2 * (c + 1) + 1] };
                v4f qq = q[c >> 1];
                v2f bv = { qq.z, qq.w };
                acc1 = __builtin_amdgcn_wmma_f32_16x16x4_f32(
                    false, av, false, bv, (short)0, acc1, false, false);
            }
        }

        const int cand = nt * 16 + lm;              // this lane's code column
        #pragma unroll
        for (int j = 0; j < 8; ++j) {
            float dot = acc0[j] + acc1[j];
            float s = fmaf(-2.0f, dot, en);         // d2 - xnorm (xnorm added later)
            if (s < minv[j]) { minv[j] = s; mini[j] = cand; }   // strict < keeps first min
        }
    }

    // ---- Butterfly min/argmin across the 16 lanes of each half ----
    #pragma unroll
    for (int m = 1; m <= 8; m <<= 1) {
        #pragma unroll
        for (int j = 0; j < 8; ++j) {
            float ov = __shfl_xor(minv[j], m, 32);
            int   oi = __shfl_xor(mini[j], m, 32);
            if (ov < minv[j] || (ov == minv[j] && oi < mini[j])) { minv[j] = ov; mini[j] = oi; }
        }
    }

    // Writer lane for row r = lm is the one whose half matches (lm>>3); slot j = lm&7.
    const bool writer = ((lm >> 3) == half);
    int jsel = lm & 7;
    float mv = minv[0]; int mi = mini[0];
    #pragma unroll
    for (int q2 = 1; q2 < 8; ++q2)
        if (jsel == q2) { mv = minv[q2]; mi = mini[q2]; }

    // Wave-level loss partial (min d2 = mv + xnorm), one atomic per wave.
    float lv = writer ? (mv + xnorm) : 0.0f;
    #pragma unroll
    for (int m = 1; m < 32; m <<= 1) lv += __shfl_xor(lv, m, 32);
    if (lane == 0) atomicAdd(loss_sum, lv);

    if (writer) {
        int n = tile * 16 + lm;
        out_idx_f[n] = (float)mi;
        atomicAdd(&counts[mi], 1.0f);
    }
}

// Scalars: commitment loss, perplexity, usage.
__global__ __launch_bounds__(512) void vq_finalize(const float* __restrict__ counts,
                                                   const float* __restrict__ loss_sum,
                                                   const float* __restrict__ weight,
                                                   float* __restrict__ out_scalars) {
    __shared__ float shH[512];
    __shared__ float shU[512];
    int t = threadIdx.x;
    float c = counts[t];
    float p = c * (1.0f / (float)N_TOT);
    shH[t] = -p * logf(p + 1e-10f);
    shU[t] = (weight[t] >= 0.01f) ? 1.0f : 0.0f;
    __syncthreads();
    for (int s = 256; s > 0; s >>= 1) {
        if (t < s) { shH[t] += shH[t + s]; shU[t] += shU[t + s]; }
        __syncthreads();
    }
    if (t == 0) {
        out_scalars[0] = loss_sum[0] * (1.0f / ((float)N_TOT * (float)DDIM));
        out_scalars[1] = expf(shH[0]);
        out_scalars[2] = shU[0];
    }
}

// quantized[b][c][h][w] = embedding[idx[n]][c], n = b*HW + h*W + w.
__global__ __launch_bounds__(256) void vq_scatter(const float* __restrict__ emb,
                                                  const float* __restrict__ idxf,
                                                  float* __restrict__ out) {
    int n = blockIdx.x * 256 + threadIdx.x;         // 512 blocks
    int k = (int)idxf[n];
    const float* er = emb + (size_t)k * DDIM;
    size_t base = (size_t)(n >> 12) * CHW + (n & (HW - 1));
    #pragma unroll 8
    for (int c = 0; c < DDIM; ++c)
        out[base + (size_t)c * HW] = er[c];         // coalesced: consecutive n -> consecutive w
}

extern "C" void kernel_launch(void* const* d_in, const int* in_sizes, int n_in,
                              void* d_out, int out_size, void* d_ws, size_t ws_size,
                              hipStream_t stream) {
    const float* x      = (const float*)d_in[0];   // [32,64,64,64] fp32
    const float* emb    = (const float*)d_in[1];   // [512,64] fp32
    const float* weight = (const float*)d_in[2];   // [512] fp32
    float* out = (float*)d_out;
    float* ws  = (float*)d_ws;

    float* Bpack  = ws;
    float* enorm  = ws + 32768;
    float* counts = ws + 33280;
    float* loss   = ws + 33792;

    float* out_scalars = out + 8388608;
    float* out_idx     = out + 8388611;

    vq_prep    <<<128,  256, 0, stream>>>(emb, Bpack, enorm, counts, loss);
    vq_main    <<<1024, 256, 0, stream>>>(x, Bpack, enorm, counts, loss, out_idx);
    vq_finalize<<<1,    512, 0, stream>>>(counts, loss, weight, out_scalars);
    vq_scatter <<<512,  256, 0, stream>>>(emb, out_idx, out);
}